// SimpleGCN_55679956025854
// MI455X (gfx1250) — compile-verified
//
#include <hip/hip_runtime.h>

typedef __attribute__((ext_vector_type(2))) float v2f;
typedef __attribute__((ext_vector_type(8))) float v8f;

// ---------------- degree / normalization ----------------

__global__ __launch_bounds__(256) void k_deg_init(float* __restrict__ deg, int N) {
    int i = blockIdx.x * blockDim.x + threadIdx.x;
    if (i < N) deg[i] = 1.0f;  // self loop
}

__global__ __launch_bounds__(256) void k_deg_count(const int* __restrict__ dst,
                                                   float* __restrict__ deg, int E) {
    int e = blockIdx.x * blockDim.x + threadIdx.x;
    if (e < E) unsafeAtomicAdd(&deg[dst[e]], 1.0f);
}

__global__ __launch_bounds__(256) void k_rsqrt(float* __restrict__ deg, int N) {
    int i = blockIdx.x * blockDim.x + threadIdx.x;
    if (i < N) deg[i] = rsqrtf(deg[i]);  // deg >= 1 always
}

// ---------------- propagation hop ----------------
// hout[i] = dinv[i]^2 * hin[i]   (self-loop term), then scatter-add edges.

__global__ __launch_bounds__(256) void k_hop_init(const float* __restrict__ hin,
                                                  const float* __restrict__ dinv,
                                                  float* __restrict__ hout, int N) {
    int tid = blockIdx.x * blockDim.x + threadIdx.x;  // N*16 threads
    if (tid >= N * 16) return;
    int node = tid >> 4;
    int c = (tid & 15) * 4;
    float s = dinv[node];
    s = s * s;
    float4 v = *(const float4*)(hin + (size_t)node * 64 + c);
    float4 o = make_float4(v.x * s, v.y * s, v.z * s, v.w * s);
    *(float4*)(hout + (size_t)node * 64 + c) = o;
}

__global__ __launch_bounds__(256) void k_hop_scatter(const float* __restrict__ hin,
                                                     const float* __restrict__ dinv,
                                                     const int* __restrict__ src,
                                                     const int* __restrict__ dst,
                                                     float* __restrict__ hout, int E) {
    int tid = blockIdx.x * blockDim.x + threadIdx.x;  // E*16 threads
    if (tid >= E * 16) return;
    int e = tid >> 4;
    int c = (tid & 15) * 4;
    int s = src[e];
    int d = dst[e];
    float norm = dinv[s] * dinv[d];
    float4 v = *(const float4*)(hin + (size_t)s * 64 + c);
    float* op = hout + (size_t)d * 64 + c;
    unsafeAtomicAdd(op + 0, v.x * norm);
    unsafeAtomicAdd(op + 1, v.y * norm);
    unsafeAtomicAdd(op + 2, v.z * norm);
    unsafeAtomicAdd(op + 3, v.w * norm);
}

// ---------------- WMMA GEMM: out = h @ W^T + bias ----------------
// h: [N,64] f32 row-major, W: [64,64] f32 (out_feat major), out: [N,64].
// One wave per 16x16 output tile; K=64 -> 16 x V_WMMA_F32_16X16X4_F32.

__global__ __launch_bounds__(256) void k_gemm_wmma(const float* __restrict__ h,
                                                   const float* __restrict__ W,
                                                   const float* __restrict__ bias,
                                                   float* __restrict__ out,
                                                   int N, int n_tiles) {
    int wave = (blockIdx.x * blockDim.x + threadIdx.x) >> 5;
    int lane = threadIdx.x & 31;
    if (wave >= n_tiles) return;  // wave-uniform: EXEC stays all-ones for WMMA

    int row_tile = wave >> 2;     // 4 col tiles (64/16)
    int col_tile = wave & 3;
    int row_base = row_tile * 16;
    int col_base = col_tile * 16;

    int mn = lane & 15;               // A: row M; B: col N
    int khalf = (lane >> 4) * 2;      // K sub-pair: 0 or 2

    int arow = row_base + mn;
    if (arow > N - 1) arow = N - 1;   // clamp for safe tail loads
    const float* ap = h + (size_t)arow * 64;
    const float* bp = W + (size_t)(col_base + mn) * 64;  // B[k][n] = W[n][k]

    v8f c = {};
#pragma unroll
    for (int kk = 0; kk < 16; ++kk) {
        int k0 = kk * 4 + khalf;
        v2f a = *(const v2f*)(ap + k0);   // A[m][k0], A[m][k0+1]
        v2f b = *(const v2f*)(bp + k0);   // B[k0][n], B[k0+1][n]
        c = __builtin_amdgcn_wmma_f32_16x16x4_f32(
                /*neg_a=*/false, a, /*neg_b=*/false, b,
                /*c_mod=*/(short)0, c, /*reuse_a=*/false, /*reuse_b=*/false);
    }

    // C/D layout: VGPR r -> M = r + (lane>>4)*8, N = lane&15
    int n = lane & 15;
    float bv = bias[col_base + n];
    int mhi = (lane >> 4) * 8;
#pragma unroll
    for (int r = 0; r < 8; ++r) {
        int m = row_base + mhi + r;
        if (m < N) out[(size_t)m * 64 + col_base + n] = c[r] + bv;
    }
}

// ---------------- launch ----------------

extern "C" void kernel_launch(void* const* d_in, const int* in_sizes, int n_in,
                              void* d_out, int out_size, void* d_ws, size_t ws_size,
                              hipStream_t stream) {
    const float* x    = (const float*)d_in[0];   // [N,64]
    const int*   ei   = (const int*)d_in[1];     // [2,E]
    // d_in[2] = edge_weights: unused by reference (SGConv drops them)
    const float* W    = (const float*)d_in[3];   // [64,64]
    const float* bias = (const float*)d_in[4];   // [64]

    int N = in_sizes[0] / 64;
    int E = in_sizes[2];                         // edge_weights count == E
    const int* src = ei;
    const int* dst = ei + E;

    float* ws   = (float*)d_ws;
    size_t npad = ((size_t)N + 63) & ~(size_t)63;
    float* dinv = ws;                 // N floats (deg, then rsqrt in place)
    float* h1   = ws + npad;          // N*64 floats
    float* h2   = h1 + (size_t)N * 64;
    float* out  = (float*)d_out;

    const int B = 256;
    int gN  = (N + B - 1) / B;
    int gE  = (E + B - 1) / B;
    int gNf = (N * 16 + B - 1) / B;
    int gEf = (E * 16 + B - 1) / B;

    // degrees -> dinv
    k_deg_init<<<gN, B, 0, stream>>>(dinv, N);
    k_deg_count<<<gE, B, 0, stream>>>(dst, dinv, E);
    k_rsqrt<<<gN, B, 0, stream>>>(dinv, N);

    // hop 1: x -> h1
    k_hop_init<<<gNf, B, 0, stream>>>(x, dinv, h1, N);
    k_hop_scatter<<<gEf, B, 0, stream>>>(x, dinv, src, dst, h1, E);

    // hop 2: h1 -> h2
    k_hop_init<<<gNf, B, 0, stream>>>(h1, dinv, h2, N);
    k_hop_scatter<<<gEf, B, 0, stream>>>(h1, dinv, src, dst, h2, E);

    // out = h2 @ W^T + bias  (WMMA f32)
    int n_tiles = ((N + 15) / 16) * 4;
    int waves_per_block = B / 32;
    int gG = (n_tiles + waves_per_block - 1) / waves_per_block;
    k_gemm_wmma<<<gG, B, 0, stream>>>(h2, W, bias, out, N, n_tiles);
}